// DualMAR_59468117180439
// MI455X (gfx1250) — compile-verified
//
#include <hip/hip_runtime.h>
#include <hip/hip_bf16.h>

#define N_NODES 100000
#define N_EDGES 3200000
#define DIM     256
#define NB      64
#define NV      50
#define NC      64
#define DEC_H1  512
#define DEC_H2  256
#define N_CLS   1000
#define NEG_INF -1e9f
#define KMAX    512            // max K any GEMM uses (Wd2 panel)

typedef __attribute__((ext_vector_type(2))) float v2f;
typedef __attribute__((ext_vector_type(8))) float v8f;
typedef int v4i __attribute__((vector_size(16)));
typedef __attribute__((address_space(1))) v4i* gptr_v4i;   // global int4*
typedef __attribute__((address_space(3))) v4i* lptr_v4i;   // LDS int4*

// ---- CDNA5 async global->LDS staging (guarded; falls back to plain copy) ----
#if defined(__has_builtin)
#if __has_builtin(__builtin_amdgcn_global_load_async_to_lds_b128)
#define HAVE_ASYNC_LDS 1
#endif
#endif

__device__ __forceinline__ void stage16B(const float* g, float* l) {
#ifdef HAVE_ASYNC_LDS
  __builtin_amdgcn_global_load_async_to_lds_b128(
      (gptr_v4i)(unsigned long long)g,
      (lptr_v4i)(unsigned int)(unsigned long long)l,
      0, 0);
#else
  *(float4*)l = *(const float4*)g;
#endif
}

__device__ __forceinline__ void wait_async_lds() {
#ifdef HAVE_ASYNC_LDS
#if __has_builtin(__builtin_amdgcn_s_wait_asynccnt)
  __builtin_amdgcn_s_wait_asynccnt(0);
#else
  asm volatile("s_wait_asynccnt 0" ::: "memory");
#endif
#endif
}

// ---------------------------------------------------------------- zero fill
__global__ void zero_f32(float* __restrict__ p, long n) {
  long i = (long)blockIdx.x * blockDim.x + threadIdx.x;
  long stride = (long)gridDim.x * blockDim.x;
  for (; i < n; i += stride) p[i] = 0.f;
}

// ------------------------------------------- weighted edge scatter-add (agg)
// out[dst[e], :] += w[e] * x[src[e], :]   (D = 256, 32 lanes per edge)
__global__ __launch_bounds__(256) void edge_agg(
    const int* __restrict__ src, const int* __restrict__ dst,
    const float* __restrict__ w, const float* __restrict__ x,
    float* __restrict__ out, int nE)
{
  long gid = (long)blockIdx.x * blockDim.x + threadIdx.x;
  int e    = (int)(gid >> 5);
  int lane = (int)(gid & 31);
  if (e >= nE) return;
  int s = src[e], d = dst[e];
  float we = w[e];
  const float4* xs = (const float4*)(x + (size_t)s * DIM);
  float* od = out + (size_t)d * DIM;
#pragma unroll
  for (int i = 0; i < 2; ++i) {
    int q = lane + 32 * i;            // float4 index 0..63
    float4 v = xs[q];
    int base = q * 4;
    atomicAdd(od + base + 0, we * v.x);
    atomicAdd(od + base + 1, we * v.y);
    atomicAdd(od + base + 2, we * v.z);
    atomicAdd(od + base + 3, we * v.w);
  }
}

// ---------------------------------------------- WMMA f32 GEMM + bias (+relu)
// Block = 8 waves = one 128x16 output tile. The K x 16 panel of W is staged
// into LDS once per block via global_load_async_to_lds_b128; each wave then
// runs V_WMMA_F32_16X16X4_F32 with A from global (b64, immediate offsets) and
// B from LDS (fused ds_load_2addr_b32, pointer-increment addressing).
// Full-M tiles take the unscaled fast path (no per-WMMA VALU overhead).
__global__ __launch_bounds__(256) void wmma_gemm(
    const float* __restrict__ A, const float* __restrict__ W,
    const float* __restrict__ bias, float* __restrict__ Cout,
    int M, int N, int K, int relu)
{
  __shared__ float wlds[KMAX][16];           // 32 KB max
  int tilesN = (N + 15) >> 4;
  int bn = blockIdx.x % tilesN;
  int bm = blockIdx.x / tilesN;
  int n0 = bn << 4;

  // ---- stage W[0:K][n0:n0+16] into LDS (16B quads, coalesced) ----
  int quads = K * 4;
  for (int q = threadIdx.x; q < quads; q += 256) {
    int k  = q >> 2;
    int n4 = (q & 3) << 2;
    const float* gsrc = W + (size_t)k * N + n0 + n4;
    float* ldst = &wlds[k][n4];
    if (n0 + n4 + 3 < N) {
      stage16B(gsrc, ldst);
    } else {
#pragma unroll
      for (int j = 0; j < 4; ++j)
        ldst[j] = (n0 + n4 + j < N) ? gsrc[j] : 0.f;
    }
  }
  wait_async_lds();
  __syncthreads();

  int lane = threadIdx.x & 31;
  int wave = threadIdx.x >> 5;
  int m0   = bm * 128 + wave * 16;
  if (m0 >= M) return;                       // wave-uniform, after barrier

  int r16  = lane & 15;                      // row (A) / col (B,C)
  int kk   = (lane >> 4) << 1;               // K sub-pair: 0 or 2
  int mrow = m0 + r16;
  const float* bp = &wlds[kk][r16];          // advances 4 rows (64 floats)/step
  v8f acc = {};

  if (m0 + 16 <= M) {
    // ---- fast path: full 16-row tile, no A masking ----
    const float* arow = A + (size_t)mrow * K;
    for (int kb = 0; kb < K; kb += 4) {
      if ((kb & 63) == 0 && kb + 64 < K)
        __builtin_prefetch(arow + kb + 64, 0, 3);
      v2f av, bv;
      av.x = arow[kb + kk];
      av.y = arow[kb + kk + 1];
      bv.x = bp[0];
      bv.y = bp[16];
      bp += 64;
      acc = __builtin_amdgcn_wmma_f32_16x16x4_f32(
          false, av, false, bv, (short)0, acc, false, false);
    }
    int ncol = n0 + r16;
    int cm   = m0 + ((lane >> 4) << 3);      // +0 or +8
    if (ncol < N) {
      float bb = bias[ncol];
#pragma unroll
      for (int r = 0; r < 8; ++r) {
        float v = acc[r] + bb;
        if (relu) v = fmaxf(v, 0.f);
        Cout[(size_t)(cm + r) * N + ncol] = v;
      }
    }
  } else {
    // ---- ragged M tail: zero-scale out-of-range rows ----
    float asc = (mrow < M) ? 1.f : 0.f;
    int   mcl = (mrow < M) ? mrow : (M - 1);
    const float* arow = A + (size_t)mcl * K;
    for (int kb = 0; kb < K; kb += 4) {
      v2f av, bv;
      av.x = arow[kb + kk] * asc;
      av.y = arow[kb + kk + 1] * asc;
      bv.x = bp[0];
      bv.y = bp[16];
      bp += 64;
      acc = __builtin_amdgcn_wmma_f32_16x16x4_f32(
          false, av, false, bv, (short)0, acc, false, false);
    }
    int ncol = n0 + r16;
    int cm   = m0 + ((lane >> 4) << 3);
    if (ncol < N) {
      float bb = bias[ncol];
#pragma unroll
      for (int r = 0; r < 8; ++r) {
        int m = cm + r;
        if (m < M) {
          float v = acc[r] + bb;
          if (relu) v = fmaxf(v, 0.f);
          Cout[(size_t)m * N + ncol] = v;
        }
      }
    }
  }
}

// ----------------------------- fused per-visit code attention (1 block/visit)
// ce = emb[codes] (64x256 in LDS); ev = sum_n tanh(ce@Wv)[:,n]*uv[n] via WMMA
// (each wave owns 2 N-tiles, 4 M-accumulators reuse each B pair 4x);
// av = softmax(mask ? ev : -1e9); vemb = av @ ce.
__global__ __launch_bounds__(256) void code_attn(
    const int* __restrict__ codes, const float* __restrict__ emb,
    const float* __restrict__ Wv, const float* __restrict__ uv,
    float* __restrict__ vemb, int* __restrict__ vmask)
{
  __shared__ float ce[NC][DIM + 4];   // +4 pad: break 16-way LDS bank conflict
  __shared__ float ev[NC];
  __shared__ float suv[DIM];
  __shared__ float sm[1];
  __shared__ int   anyv;

  int visit = blockIdx.x;             // b*NV + v
  int tid   = threadIdx.x;
  int lane  = tid & 31;
  int wid   = tid >> 5;
  const int* cids = codes + (size_t)visit * NC;

  suv[tid] = uv[tid];
  {
    // 4 threads per code row, 64 floats (16 float4) each
    int r = tid >> 2, part = tid & 3;
    int cid = cids[r];                // pad id 0 gathers emb[0] like reference
    const float4* srow = (const float4*)(emb + (size_t)cid * DIM);
    float4* drow = (float4*)&ce[r][part * 64];
#pragma unroll
    for (int i = 0; i < 16; ++i) drow[i] = srow[part * 16 + i];
  }
  if (tid < NC) ev[tid] = 0.f;
  __syncthreads();

  int rowA = lane & 15;
  int kk   = (lane >> 4) << 1;
  const float* cr0 = &ce[rowA][0];
  const float* cr1 = &ce[rowA + 16][0];
  const float* cr2 = &ce[rowA + 32][0];
  const float* cr3 = &ce[rowA + 48][0];

#pragma unroll
  for (int ti = 0; ti < 2; ++ti) {
    int tn = wid + ti * 8;                           // this wave's N-tile
    const float* wp = Wv + (size_t)kk * DIM + tn * 16 + rowA;
    v8f a0 = {}, a1 = {}, a2 = {}, a3 = {};
    for (int kb = 0; kb < DIM; kb += 4, wp += 4 * DIM) {
      v2f bv;  bv.x = wp[0];  bv.y = wp[DIM];        // shared by 4 M-tiles
      v2f x0, x1, x2, x3;
      x0.x = cr0[kb + kk]; x0.y = cr0[kb + kk + 1];
      x1.x = cr1[kb + kk]; x1.y = cr1[kb + kk + 1];
      x2.x = cr2[kb + kk]; x2.y = cr2[kb + kk + 1];
      x3.x = cr3[kb + kk]; x3.y = cr3[kb + kk + 1];
      a0 = __builtin_amdgcn_wmma_f32_16x16x4_f32(false, x0, false, bv, (short)0, a0, false, false);
      a1 = __builtin_amdgcn_wmma_f32_16x16x4_f32(false, x1, false, bv, (short)0, a1, false, false);
      a2 = __builtin_amdgcn_wmma_f32_16x16x4_f32(false, x2, false, bv, (short)0, a2, false, false);
      a3 = __builtin_amdgcn_wmma_f32_16x16x4_f32(false, x3, false, bv, (short)0, a3, false, false);
    }
    float u  = suv[tn * 16 + rowA];
    int mb   = (lane >> 4) << 3;                     // +0 or +8
#pragma unroll
    for (int r = 0; r < 8; ++r) {
      atomicAdd(&ev[mb + r],      tanhf(a0[r]) * u); // ds_add_f32
      atomicAdd(&ev[16 + mb + r], tanhf(a1[r]) * u);
      atomicAdd(&ev[32 + mb + r], tanhf(a2[r]) * u);
      atomicAdd(&ev[48 + mb + r], tanhf(a3[r]) * u);
    }
  }
  __syncthreads();

  if (tid == 0) {     // masked softmax over 64 codes (tiny, serial)
    float mx = -3.0e38f; int av = 0;
    for (int c = 0; c < NC; ++c) {
      float e = (cids[c] > 0) ? ev[c] : NEG_INF;
      ev[c] = e;
      if (e > mx) mx = e;
      if (cids[c] > 0) av = 1;
    }
    float s = 0.f;
    for (int c = 0; c < NC; ++c) { float w = __expf(ev[c] - mx); ev[c] = w; s += w; }
    sm[0] = 1.f / s;
    anyv = av;
  }
  __syncthreads();

  float acc = 0.f;
  for (int c = 0; c < NC; ++c) acc += ev[c] * ce[c][tid];
  acc *= sm[0];
  vemb[(size_t)visit * DIM + tid] = anyv ? acc : 0.f;
  if (tid == 0) vmask[visit] = anyv;
}

// --------------------------- per-patient visit attention (1 block/patient)
__global__ __launch_bounds__(256) void visit_attn(
    const float* __restrict__ vemb, const int* __restrict__ vmask,
    const float* __restrict__ Wp, const float* __restrict__ up,
    float* __restrict__ pemb)
{
  __shared__ float sred[256];
  __shared__ float ap[NV];
  __shared__ float sm[1];
  __shared__ int   anyv;
  int b = blockIdx.x, tid = threadIdx.x;
  const float* vb = vemb + (size_t)b * NV * DIM;

  for (int v = 0; v < NV; ++v) {
    const float* vr = vb + v * DIM;
    float t = 0.f;
    for (int d = 0; d < DIM; ++d) t += vr[d] * Wp[(size_t)d * DIM + tid];
    sred[tid] = tanhf(t) * up[tid];
    __syncthreads();
    for (int s = 128; s > 0; s >>= 1) {
      if (tid < s) sred[tid] += sred[tid + s];
      __syncthreads();
    }
    if (tid == 0) ap[v] = sred[0];
    __syncthreads();
  }
  if (tid == 0) {
    float mx = -3.0e38f; int av = 0;
    for (int v = 0; v < NV; ++v) {
      float e = vmask[b * NV + v] ? ap[v] : NEG_INF;
      ap[v] = e; if (e > mx) mx = e;
      if (vmask[b * NV + v]) av = 1;
    }
    float s = 0.f;
    for (int v = 0; v < NV; ++v) { float w = __expf(ap[v] - mx); ap[v] = w; s += w; }
    sm[0] = 1.f / s;
    anyv = av;
  }
  __syncthreads();
  float inv = sm[0], acc = 0.f;
  for (int v = 0; v < NV; ++v) acc += ap[v] * vb[v * DIM + tid];
  pemb[(size_t)b * DIM + tid] = anyv ? acc * inv : 0.f;
}

// ----------------------------------------------------------------- launcher
extern "C" void kernel_launch(void* const* d_in, const int* in_sizes, int n_in,
                              void* d_out, int out_size, void* d_ws, size_t ws_size,
                              hipStream_t stream) {
  const int*   edge_index  = (const int*)d_in[0];
  const float* edge_weight = (const float*)d_in[1];
  const int*   codes       = (const int*)d_in[2];
  const float* node_emb    = (const float*)d_in[3];
  const float* W1  = (const float*)d_in[4];
  const float* b1  = (const float*)d_in[5];
  const float* W2  = (const float*)d_in[6];
  const float* b2  = (const float*)d_in[7];
  const float* Wv  = (const float*)d_in[8];
  const float* uv  = (const float*)d_in[9];
  const float* Wp  = (const float*)d_in[10];
  const float* up  = (const float*)d_in[11];
  const float* Wd1 = (const float*)d_in[12];
  const float* bd1 = (const float*)d_in[13];
  const float* Wd2 = (const float*)d_in[14];
  const float* bd2 = (const float*)d_in[15];
  const float* Wd3 = (const float*)d_in[16];
  const float* bd3 = (const float*)d_in[17];
  float* out = (float*)d_out;

  char* ws = (char*)d_ws;
  size_t off = 0;
  auto carve = [&](size_t bytes) -> void* {
    void* p = ws + off;
    off += (bytes + 255) & ~(size_t)255;
    return p;
  };
  float* bufA     = (float*)carve((size_t)N_NODES * DIM * 4);   // agg scratch
  float* bufB     = (float*)carve((size_t)N_NODES * DIM * 4);   // h / emb
  float* vembBuf  = (float*)carve((size_t)NB * NV * DIM * 4);
  int*   vmaskBuf = (int*)  carve((size_t)NB * NV * 4);
  float* pembBuf  = (float*)carve((size_t)NB * DIM * 4);
  float* h1Buf    = (float*)carve((size_t)NB * DEC_H1 * 4);
  float* h2Buf    = (float*)carve((size_t)NB * DEC_H2 * 4);

  const int* src = edge_index;
  const int* dst = edge_index + N_EDGES;

  long nAgg = (long)N_NODES * DIM;
  long eThreads = (long)N_EDGES * 32;
  int  eBlocks  = (int)((eThreads + 255) / 256);

  auto gemm = [&](const float* A, const float* W, const float* bias,
                  float* C, int M, int N, int K, int relu) {
    int tilesN  = (N + 15) / 16;
    int blocksM = (M + 127) / 128;               // 8 waves of 16 rows each
    hipLaunchKernelGGL(wmma_gemm, dim3(blocksM * tilesN), dim3(256), 0, stream,
                       A, W, bias, C, M, N, K, relu);
  };

  // GCN layer 1
  hipLaunchKernelGGL(zero_f32, dim3(4096), dim3(256), 0, stream, bufA, nAgg);
  hipLaunchKernelGGL(edge_agg, dim3(eBlocks), dim3(256), 0, stream,
                     src, dst, edge_weight, node_emb, bufA, N_EDGES);
  gemm(bufA, W1, b1, bufB, N_NODES, DIM, DIM, /*relu=*/1);      // h
  // GCN layer 2
  hipLaunchKernelGGL(zero_f32, dim3(4096), dim3(256), 0, stream, bufA, nAgg);
  hipLaunchKernelGGL(edge_agg, dim3(eBlocks), dim3(256), 0, stream,
                     src, dst, edge_weight, bufB, bufA, N_EDGES);
  gemm(bufA, W2, b2, bufB, N_NODES, DIM, DIM, /*relu=*/0);      // emb
  // attention stack
  hipLaunchKernelGGL(code_attn, dim3(NB * NV), dim3(256), 0, stream,
                     codes, bufB, Wv, uv, vembBuf, vmaskBuf);
  hipLaunchKernelGGL(visit_attn, dim3(NB), dim3(256), 0, stream,
                     vembBuf, vmaskBuf, Wp, up, pembBuf);
  // decoder MLP
  gemm(pembBuf, Wd1, bd1, h1Buf, NB, DEC_H1, DIM,    /*relu=*/1);
  gemm(h1Buf,   Wd2, bd2, h2Buf, NB, DEC_H2, DEC_H1, /*relu=*/1);
  gemm(h2Buf,   Wd3, bd3, out,   NB, N_CLS,  DEC_H2, /*relu=*/0);
}